// EmotionalGravityManifold_24043226923358
// MI455X (gfx1250) — compile-verified
//
#include <hip/hip_runtime.h>

typedef __attribute__((ext_vector_type(2))) float v2f;
typedef __attribute__((ext_vector_type(4))) float v4f;
typedef __attribute__((ext_vector_type(8))) float v8f;

#define PTS 4  // points per thread; one wave owns PTS*32 consecutive points

__device__ __forceinline__ float dot8(v4f a0, v4f a1, v4f b0, v4f b1) {
  return (a0.x*b0.x + a0.y*b0.y) + (a0.z*b0.z + a0.w*b0.w) +
         (a1.x*b1.x + a1.y*b1.y) + (a1.z*b1.z + a1.w*b1.w);
}

// One 4-WMMA chain covering 32 points: B-operand loads use the wave-uniform
// SGPR base wp = pos + tb*8 with small 32-bit per-lane offsets.
__device__ __forceinline__ v8f xy_chain(const float* __restrict__ wp, int m, int kb,
                                        v2f a1k0, v2f a1k1, v2f a2k0, v2f a2k1) {
  const int o1 = m * 8 + kb;
  const int o2 = (16 + m) * 8 + kb;
  v2f b1k0 = { wp[o1],     wp[o1 + 1] };
  v2f b1k1 = { wp[o1 + 4], wp[o1 + 5] };
  v2f b2k0 = { wp[o2],     wp[o2 + 1] };
  v2f b2k1 = { wp[o2 + 4], wp[o2 + 5] };
  v8f c = {0.f,0.f,0.f,0.f,0.f,0.f,0.f,0.f};
  c = __builtin_amdgcn_wmma_f32_16x16x4_f32(false, a1k0, false, b1k0, (short)0, c, false, false);
  c = __builtin_amdgcn_wmma_f32_16x16x4_f32(false, a1k1, false, b1k1, (short)0, c, false, false);
  c = __builtin_amdgcn_wmma_f32_16x16x4_f32(false, a2k0, false, b2k0, (short)0, c, false, false);
  c = __builtin_amdgcn_wmma_f32_16x16x4_f32(false, a2k1, false, b2k1, (short)0, c, false, false);
  return c;
}

// Per-point hyperbolic force: cx[n] = pos.att[n] (from WMMA), y2s[n] uniform SGPRs.
__device__ __forceinline__ void force_point(const v8f cx, const float* __restrict__ att,
                                            const float* __restrict__ str,
                                            const float* y2s,
                                            v4f x0, v4f x1, v4f& acc0, v4f& acc1) {
  const float x2 = dot8(x0, x1, x0, x1);
  const float oneMinusX2 = 1.0f - x2;            // (1 - c*x2), c = 1
  const float kk = oneMinusX2 + 1e-8f;           // 2/lam

  acc0 = (v4f){0.f,0.f,0.f,0.f};
  acc1 = (v4f){0.f,0.f,0.f,0.f};

#pragma unroll
  for (int n = 0; n < 6; ++n) {
    const v4f y0 = *(const v4f*)(att + n * 8);   // uniform -> SGPRs
    const v4f y1 = *(const v4f*)(att + n * 8 + 4);
    const float y2 = y2s[n];
    const float pa = cx[n];

    // mobius_add(-pos, att): xy = -pa
    const float t2    = 1.0f - 2.0f * pa;
    const float coefA = t2 + y2;                 // (1 + 2c*xy + c*y2)
    const float denom = fmaf(x2, y2, t2) + 1e-8f;
    const float rd    = __builtin_amdgcn_rcpf(denom);
    const float ca    = coefA * rd;
    const float cb    = oneMinusX2 * rd;

    const v4f d0 = y0 * cb - x0 * ca;            // diff
    const v4f d1 = y1 * cb - x1 * ca;

    const float nrm2 = dot8(d0, d1, d0, d1);
    const float rsq  = __builtin_amdgcn_rsqf(nrm2);   // 1/|diff|
    const float dn   = nrm2 * rsq;                    // |diff|
    const float dnc  = fminf(fmaxf(dn, 1e-8f), 0.99999f);

    // atanh(z) = 0.5*ln2 * log2((1+z)/(1-z))
    const float at = 0.346573590279973f *
        __builtin_amdgcn_logf((1.0f + dnc) * __builtin_amdgcn_rcpf(1.0f - dnc));

    const float dist = 2.0f * at;
    const float fm   = str[n] * __builtin_amdgcn_rcpf(fmaf(dist, dist, 0.1f));

    // direction = s*diff, s > 0 => normalized branch is diff*(fm/dn)
    const float s = kk * at * __builtin_amdgcn_rcpf(dnc);
    const float w = (s * dn > 1e-6f) ? (fm * rsq) : (fm * s);

    acc0 += d0 * w;
    acc1 += d1 * w;
  }
}

__global__ __launch_bounds__(256) void egm_kernel(
    const float* __restrict__ pos, const float* __restrict__ att,
    const float* __restrict__ str, float* __restrict__ out, int B)
{
  const int lane = (int)threadIdx.x & 31;
  const int m    = lane & 15;
  const int kb   = (lane < 16) ? 0 : 2;

  // Wave-uniform by construction; readfirstlane tells the compiler, so the
  // chain bases stay in SGPRs (s_min_i32 + saddr-form loads, no 64-bit VALU).
  const int waveBase = __builtin_amdgcn_readfirstlane(
      (blockIdx.x * 256 + ((int)threadIdx.x & ~31)) * PTS);

  // ---- A operands: 16x4 attractor tiles (rows 0..5 and rows 8..13).
  v2f a1k0 = {0.f,0.f}, a1k1 = {0.f,0.f};
  v2f a2k0 = {0.f,0.f}, a2k1 = {0.f,0.f};
  if (m < 6) {
    const float* ap = att + m * 8 + kb;
    a1k0.x = ap[0]; a1k0.y = ap[1];
    a1k1.x = ap[4]; a1k1.y = ap[5];
  }
  if (m >= 8 && m < 14) {
    const float* ap = att + (m - 8) * 8 + kb;
    a2k0.x = ap[0]; a2k0.y = ap[1];
    a2k1.x = ap[4]; a2k1.y = ap[5];
  }

  // ---- Attractor Gram matrix on the matrix unit: a1k0 is bit-identical to
  // the B-operand layout for "columns = attractors", so G = att @ att^T and
  // y2[n] = G[n][n] lives at (VGPR n, lane n) -> v_readlane into SGPRs.
  v8f g = {0.f,0.f,0.f,0.f,0.f,0.f,0.f,0.f};
  g = __builtin_amdgcn_wmma_f32_16x16x4_f32(false, a1k0, false, a1k0, (short)0, g, false, false);
  g = __builtin_amdgcn_wmma_f32_16x16x4_f32(false, a1k1, false, a1k1, (short)0, g, false, false);
  float y2s[6];
#pragma unroll
  for (int n = 0; n < 6; ++n)
    y2s[n] = __int_as_float(__builtin_amdgcn_readlane(__float_as_int(g[n]), n));

  // ---- Single code path: clamp each 32-point chain BASE (scalar min).
  // A tail wave recomputes points another wave also computes — identical
  // inputs give identical outputs, so overlapping stores are benign, all
  // accesses stay in bounds, and EXEC stays all-ones for the WMMAs.
  // (Assumes B >= 32; here B = 1048576 and the tile divides B exactly.)
#pragma unroll
  for (int p = 0; p < PTS; ++p) {
    const int tb = min(waveBase + p * 32, B - 32);    // SALU
    const float* wp = pos + (size_t)tb * 8;           // SGPR base
    const v8f cx = xy_chain(wp, m, kb, a1k0, a1k1, a2k0, a2k1);

    const v4f x0 = *(const v4f*)(wp + lane * 8);
    const v4f x1 = *(const v4f*)(wp + lane * 8 + 4);
    v4f acc0, acc1;
    force_point(cx, att, str, y2s, x0, x1, acc0, acc1);

    // Output is write-once: non-temporal stores keep it out of the caches.
    float* wo = out + (size_t)tb * 8;                 // SGPR base
    __builtin_nontemporal_store(acc0, (v4f*)(wo + lane * 8));
    __builtin_nontemporal_store(acc1, (v4f*)(wo + lane * 8 + 4));
  }
}

extern "C" void kernel_launch(void* const* d_in, const int* in_sizes, int n_in,
                              void* d_out, int out_size, void* d_ws, size_t ws_size,
                              hipStream_t stream) {
  const float* pos = (const float*)d_in[0];   // (B, 8) float32
  const float* att = (const float*)d_in[1];   // (6, 8) float32
  const float* str = (const float*)d_in[2];   // (6,)   float32
  float* out = (float*)d_out;                 // (B, 8) float32

  const int B = in_sizes[0] / 8;
  const int ptsPerBlock = 256 * PTS;          // 1024 points per block
  const int blocks = (B + ptsPerBlock - 1) / ptsPerBlock;
  egm_kernel<<<blocks, 256, 0, stream>>>(pos, att, str, out, B);
}